// GVPTransformerLayer_8332236554742
// MI455X (gfx1250) — compile-verified
//
#include <hip/hip_runtime.h>

#define Nn 20000
#define Ee 160000
#define Bb 64
#define Hh 8
#define EPSF 1e-8f
#define ENC_NINF 0x007FFFFFu  /* fenc(-inf) */
#define ENC_PINF 0xFF800000u  /* fenc(+inf) */

typedef float v2f __attribute__((ext_vector_type(2)));
typedef float v8f __attribute__((ext_vector_type(8)));

__device__ __forceinline__ unsigned fenc(float f) {
  unsigned u = __float_as_uint(f);
  return (u & 0x80000000u) ? ~u : (u | 0x80000000u);
}
__device__ __forceinline__ float fdec(unsigned u) {
  return __uint_as_float((u & 0x80000000u) ? (u & 0x7fffffffu) : ~u);
}

// -------------------- fills --------------------
__global__ void fill_f32_kernel(float* p, float v, long long n) {
  long long i = (long long)blockIdx.x * blockDim.x + threadIdx.x;
  if (i < n) p[i] = v;
}
__global__ void fill_u32_kernel(unsigned* p, unsigned v, long long n) {
  long long i = (long long)blockIdx.x * blockDim.x + threadIdx.x;
  if (i < n) p[i] = v;
}

// -------------------- WMMA fp32 linear: out[M][Nout] = A[M][K] @ W[Nout][K]^T (+bias) ---
// One wave32 per 16x16 output tile; K-loop of V_WMMA_F32_16X16X4_F32 with D->C chaining.
__global__ __launch_bounds__(32)
void lin_wmma_kernel(float* __restrict__ out, const float* __restrict__ A,
                     const float* __restrict__ W, const float* __restrict__ bias,
                     int M, int K, int Nout) {
  int row0 = blockIdx.x * 16;
  int col0 = blockIdx.y * 16;
  int l = threadIdx.x;
  int lm = l & 15;
  int koff = (l >> 4) * 2;       // lanes 16-31 hold K=+2,+3 of the fragment
  const float* arow = A + (size_t)(row0 + lm) * K;
  const float* wrow = W + (size_t)(col0 + lm) * K;
  v8f acc = {};
  for (int k = 0; k < K; k += 4) {
    v2f a, b;
    a.x = arow[k + koff];  a.y = arow[k + koff + 1];
    b.x = wrow[k + koff];  b.y = wrow[k + koff + 1];
    acc = __builtin_amdgcn_wmma_f32_16x16x4_f32(false, a, false, b, (short)0, acc,
                                                false, false);
  }
  float bsv = bias ? bias[col0 + lm] : 0.0f;
  int rbase = row0 + ((l >> 4) ? 8 : 0);   // D: VGPR r = row r (lanes 0-15) / r+8 (16-31)
#pragma unroll
  for (int r = 0; r < 8; ++r)
    out[(size_t)(rbase + r) * Nout + col0 + lm] = acc[r] + bsv;
}

// -------------------- WMMA vector-channel einsum: out[n][o][c] = sum_v W[o][v]*X[n][v][c]
// Columns = flattened (n*3+c); grid.x covers 3*NN/16 (always a multiple of 16 here).
__global__ __launch_bounds__(32)
void vlin_wmma_kernel(float* __restrict__ out, const float* __restrict__ X,
                      const float* __restrict__ W, int O, int V) {
  int col0 = blockIdx.x * 16;
  int row0 = blockIdx.y * 16;
  int l = threadIdx.x;
  int lm = l & 15;
  int koff = (l >> 4) * 2;
  int f = col0 + lm;
  int n = f / 3, c = f % 3;
  int ro = row0 + lm;
  float rmask = (ro < O) ? 1.0f : 0.0f;   // pad partial O tiles with zero rows
  int roc = (ro < O) ? ro : 0;
  const float* wr = W + (size_t)roc * V;
  v8f acc = {};
  for (int k = 0; k < V; k += 4) {
    int kk = k + koff;
    v2f a, b;
    a.x = wr[kk] * rmask;  a.y = wr[kk + 1] * rmask;
    b.x = X[((size_t)n * V + kk) * 3 + c];
    b.y = X[((size_t)n * V + kk + 1) * 3 + c];
    acc = __builtin_amdgcn_wmma_f32_16x16x4_f32(false, a, false, b, (short)0, acc,
                                                false, false);
  }
  int rbase = row0 + ((l >> 4) ? 8 : 0);
#pragma unroll
  for (int r = 0; r < 8; ++r) {
    int o = rbase + r;
    if (o < O) out[((size_t)n * O + o) * 3 + c] = acc[r];
  }
}

// -------------------- attention --------------------
__global__ void attn_logits_kernel(const int* __restrict__ ei,
                                   const float* __restrict__ qs, const float* __restrict__ ks,
                                   const float* __restrict__ qv, const float* __restrict__ kv,
                                   const float* __restrict__ be,
                                   float* __restrict__ as_, float* __restrict__ av_,
                                   float* __restrict__ ae_, float* __restrict__ lg_,
                                   unsigned* __restrict__ lmax) {
  int i = blockIdx.x * blockDim.x + threadIdx.x;
  if (i >= Ee * Hh) return;
  int e = i >> 3, h = i & 7;
  int r = ei[e], c = ei[Ee + e];
  const float* q = qs + (size_t)r * 64 + h * 8;
  const float* k = ks + (size_t)c * 64 + h * 8;
  float as = 0.f;
#pragma unroll
  for (int d = 0; d < 8; ++d) as += q[d] * k[d];
  const float* qp = qv + ((size_t)r * 16 + h * 2) * 3;
  const float* kp = kv + ((size_t)c * 16 + h * 2) * 3;
  float av = 0.f;
#pragma unroll
  for (int d = 0; d < 6; ++d) av += qp[d] * kp[d];
  const float* bp = be + (size_t)e * 32 + h * 4;
  float ae = bp[0] + bp[1] + bp[2] + bp[3];
  as_[i] = as; av_[i] = av; ae_[i] = ae;
  float lg = as * 0.20412414523193154f     // 1/sqrt(3*DKS)
           + av * 0.23570226039551587f     // 1/sqrt(9*DKV)
           + ae * 0.28867513459481287f;    // 1/sqrt(3*DB)
  lg_[i] = lg;
  atomicMax(&lmax[(size_t)r * 8 + h], fenc(lg));
}

__global__ void attn_exp_kernel(const int* __restrict__ ei, float* __restrict__ at,
                                const unsigned* __restrict__ lmax, float* __restrict__ den) {
  int i = blockIdx.x * blockDim.x + threadIdx.x;
  if (i >= Ee * Hh) return;
  int e = i >> 3, h = i & 7;
  int r = ei[e];
  float ex = expf(at[i] - fdec(lmax[(size_t)r * 8 + h]));
  at[i] = ex;
  atomicAdd(&den[(size_t)r * 8 + h], ex);
}

__global__ void attn_norm_kernel(const int* __restrict__ ei, float* __restrict__ at,
                                 const float* __restrict__ den) {
  int i = blockIdx.x * blockDim.x + threadIdx.x;
  if (i >= Ee * Hh) return;
  int e = i >> 3, h = i & 7;
  at[i] = at[i] / den[(size_t)ei[e] * 8 + h];
}

__global__ void count_edges_kernel(const int* __restrict__ ei, float* __restrict__ cnt) {
  int i = blockIdx.x * blockDim.x + threadIdx.x;
  if (i >= Ee) return;
  atomicAdd(&cnt[ei[i]], 1.0f);
}

// -------------------- message build / aggregation --------------------
__global__ void build_msg_kernel(const int* __restrict__ ei, const float* __restrict__ at,
                                 const float* __restrict__ vxs, const float* __restrict__ ves,
                                 const float* __restrict__ vxv, const float* __restrict__ vev,
                                 float* __restrict__ mcat, float* __restrict__ mvp) {
  int i = blockIdx.x * blockDim.x + threadIdx.x;
  if (i >= Ee * Hh) return;
  int e = i >> 3, h = i & 7;
  int c = ei[Ee + e];
  float a = at[i];
  const float* vs = vxs + (size_t)c * 64 + h * 8;
  const float* vp = ves + (size_t)e * 64 + h * 8;
  float* mc = mcat + (size_t)e * 160 + h * 16;
#pragma unroll
  for (int d = 0; d < 8; ++d) { mc[d] = vs[d] * a; mc[8 + d] = vp[d] * a; }
  const float* xv = vxv + ((size_t)c * 16 + h * 2) * 3;
  const float* ep = vev + ((size_t)e * 16 + h * 2) * 3;
  float* mp = mvp + ((size_t)e * 32 + h * 4) * 3;
#pragma unroll
  for (int j = 0; j < 6; ++j) { mp[j] = xv[j] * a; mp[6 + j] = ep[j] * a; }
}

// out[n][S+O] = [s_in[n][S] | norm3(vh[n][o])]  (s_in may alias out)
__global__ void concat_vn_kernel(float* __restrict__ out, const float* __restrict__ sin,
                                 int S, const float* __restrict__ vh, int O, long long NN) {
  long long i = (long long)blockIdx.x * blockDim.x + threadIdx.x;
  int T = S + O;
  if (i >= NN * T) return;
  long long n = i / T; int j = (int)(i % T);
  if (j < S) {
    out[(size_t)n * T + j] = sin[(size_t)n * S + j];
  } else {
    const float* p = vh + ((size_t)n * O + (j - S)) * 3;
    float sum = p[0] * p[0] + p[1] * p[1] + p[2] * p[2];
    out[(size_t)n * T + j] = sqrtf(fmaxf(sum, EPSF));
  }
}

__global__ void seg_accum_kernel(const int* __restrict__ ei, const float* __restrict__ mso,
                                 const float* __restrict__ mvo, float* __restrict__ xs,
                                 float* __restrict__ xv) {
  long long i = (long long)blockIdx.x * blockDim.x + threadIdx.x;
  if (i >= (long long)Ee * 224) return;
  long long e = i / 224; int j = (int)(i % 224);
  int r = ei[e];
  if (j < 128) atomicAdd(&xs[(size_t)r * 160 + j], mso[(size_t)e * 128 + j]);
  else         atomicAdd(&xv[(size_t)r * 108 + (j - 128)], mvo[(size_t)e * 96 + (j - 128)]);
}

__global__ void finish_xin_kernel(const int* __restrict__ bm, const float* __restrict__ gs,
                                  const float* __restrict__ gv, const float* __restrict__ cnt,
                                  float* __restrict__ xs, float* __restrict__ xv) {
  int n = blockIdx.x * blockDim.x + threadIdx.x;
  if (n >= Nn) return;
  float inv = 1.0f / fmaxf(cnt[n], 1.0f);
  int b = bm[n];
  float* sp = xs + (size_t)n * 160;
  for (int j = 0; j < 128; ++j) sp[j] *= inv;
  for (int j = 0; j < 32; ++j) sp[128 + j] = gs[(size_t)b * 32 + j];
  float* vp = xv + (size_t)n * 108;
  for (int j = 0; j < 96; ++j) vp[j] *= inv;
  for (int j = 0; j < 12; ++j) vp[96 + j] = gv[(size_t)b * 12 + j];
}

// -------------------- tuple LayerNorm --------------------
__global__ void tuple_ln_kernel(float* __restrict__ outs, float* __restrict__ outv,
                                const float* __restrict__ s0, const float* __restrict__ ds,
                                const float* __restrict__ v0, const float* __restrict__ dv,
                                const float* __restrict__ g, const float* __restrict__ bb,
                                const float* __restrict__ vw, int S, int V, long long NN) {
  long long n = (long long)blockIdx.x * blockDim.x + threadIdx.x;
  if (n >= NN) return;
  const float* sp = s0 + (size_t)n * S;
  const float* dp = ds + (size_t)n * S;
  float mu = 0.f;
  for (int j = 0; j < S; ++j) mu += sp[j] + dp[j];
  mu /= (float)S;
  float var = 0.f;
  for (int j = 0; j < S; ++j) { float d = sp[j] + dp[j] - mu; var += d * d; }
  var /= (float)S;
  float inv = 1.0f / sqrtf(var + 1e-5f);
  float* os = outs + (size_t)n * S;
  for (int j = 0; j < S; ++j) os[j] = (sp[j] + dp[j] - mu) * inv * g[j] + bb[j];
  const float* vp = v0 + (size_t)n * V * 3;
  const float* dq = dv + (size_t)n * V * 3;
  float acc = 0.f;
  for (int i = 0; i < V; ++i) {
    float a0 = vp[i * 3] + dq[i * 3];
    float a1 = vp[i * 3 + 1] + dq[i * 3 + 1];
    float a2 = vp[i * 3 + 2] + dq[i * 3 + 2];
    acc += fmaxf(a0 * a0 + a1 * a1 + a2 * a2, EPSF);
  }
  float dn = sqrtf(acc / (float)V);
  float* ov = outv + (size_t)n * V * 3;
  for (int i = 0; i < V; ++i) {
    float sc = vw[i] / dn;
    ov[i * 3]     = (vp[i * 3] + dq[i * 3]) * sc;
    ov[i * 3 + 1] = (vp[i * 3 + 1] + dq[i * 3 + 1]) * sc;
    ov[i * 3 + 2] = (vp[i * 3 + 2] + dq[i * 3 + 2]) * sc;
  }
}

// -------------------- edge-out build --------------------
__global__ void ebm_kernel(int* __restrict__ ebm, const int* __restrict__ ei,
                           const int* __restrict__ bm) {
  int i = blockIdx.x * blockDim.x + threadIdx.x;
  if (i >= Ee) return;
  ebm[i] = bm[ei[i]];
}

__global__ void build_es_ev_kernel(const int* __restrict__ ei, const int* __restrict__ ebm,
                                   const float* __restrict__ s, const float* __restrict__ v,
                                   const float* __restrict__ es, const float* __restrict__ ev,
                                   const float* __restrict__ gs, const float* __restrict__ gv,
                                   const float* __restrict__ as_, const float* __restrict__ av_,
                                   const float* __restrict__ ae_,
                                   float* __restrict__ e_s, float* __restrict__ e_v) {
  long long i = (long long)blockIdx.x * blockDim.x + threadIdx.x;
  if (i >= (long long)Ee * 288) return;
  long long e = i / 288; int j = (int)(i % 288);
  int r = ei[e], c = ei[Ee + e], b = ebm[e];
  if (j < 216) {
    float val;
    if (j < 64)        val = s[(size_t)r * 64 + j];
    else if (j < 128)  val = s[(size_t)c * 64 + (j - 64)];
    else if (j < 160)  val = es[(size_t)e * 32 + (j - 128)];
    else if (j < 168)  val = as_[(size_t)e * 8 + (j - 160)];
    else if (j < 176)  val = av_[(size_t)e * 8 + (j - 168)];
    else if (j < 184)  val = ae_[(size_t)e * 8 + (j - 176)];
    else               val = gs[(size_t)b * 32 + (j - 184)];
    e_s[(size_t)e * 216 + j] = val;
  } else {
    int jv = j - 216; int i2 = jv / 3, cc = jv % 3;
    float val;
    if (i2 < 8)        val = v[((size_t)r * 8 + i2) * 3 + cc];
    else if (i2 < 16)  val = v[((size_t)c * 8 + (i2 - 8)) * 3 + cc];
    else if (i2 < 20)  val = ev[((size_t)e * 4 + (i2 - 16)) * 3 + cc];
    else               val = gv[((size_t)b * 4 + (i2 - 20)) * 3 + cc];
    e_v[(size_t)e * 72 + jv] = val;
  }
}

// -------------------- PNA --------------------
__global__ void pna_accum_kernel(const float* __restrict__ s, const float* __restrict__ v,
                                 const int* __restrict__ seg,
                                 float* __restrict__ cnt, float* __restrict__ ssum,
                                 float* __restrict__ ssq, unsigned* __restrict__ smin,
                                 unsigned* __restrict__ smax, float* __restrict__ vns,
                                 float* __restrict__ vnq, unsigned* __restrict__ vnmin,
                                 unsigned* __restrict__ vnmax, float* __restrict__ vsum,
                                 int S, int V, int NN) {
  int n = blockIdx.x * blockDim.x + threadIdx.x;
  if (n >= NN) return;
  int b = seg[n];
  atomicAdd(&cnt[b], 1.0f);
  for (int j = 0; j < S; ++j) {
    float x = s[(size_t)n * S + j];
    atomicAdd(&ssum[b * S + j], x);
    atomicAdd(&ssq[b * S + j], x * x);
    atomicMin(&smin[b * S + j], fenc(x));
    atomicMax(&smax[b * S + j], fenc(x));
  }
  for (int i = 0; i < V; ++i) {
    float a0 = v[((size_t)n * V + i) * 3];
    float a1 = v[((size_t)n * V + i) * 3 + 1];
    float a2 = v[((size_t)n * V + i) * 3 + 2];
    float vn = sqrtf(fmaxf(a0 * a0 + a1 * a1 + a2 * a2, EPSF));
    atomicAdd(&vns[b * V + i], vn);
    atomicAdd(&vnq[b * V + i], vn * vn);
    atomicMin(&vnmin[b * V + i], fenc(vn));
    atomicMax(&vnmax[b * V + i], fenc(vn));
    atomicAdd(&vsum[(b * V + i) * 3], a0);
    atomicAdd(&vsum[(b * V + i) * 3 + 1], a1);
    atomicAdd(&vsum[(b * V + i) * 3 + 2], a2);
  }
}

__global__ void pna_final_kernel(float* __restrict__ z, float* __restrict__ vm,
                                 const float* __restrict__ cnt, const float* __restrict__ ssum,
                                 const float* __restrict__ ssq, const unsigned* __restrict__ smin,
                                 const unsigned* __restrict__ smax, const float* __restrict__ vns,
                                 const float* __restrict__ vnq, const unsigned* __restrict__ vnmin,
                                 const unsigned* __restrict__ vnmax, const float* __restrict__ vsum,
                                 int S, int V, int Bn) {
  int b = blockIdx.x * blockDim.x + threadIdx.x;
  if (b >= Bn) return;
  float c = fmaxf(cnt[b], 1.0f);
  float bessel = c / fmaxf(c - 1.0f, 1.0f);
  int Z = 4 * S + 3 * V;
  for (int j = 0; j < S; ++j) {
    float m = ssum[b * S + j] / c;
    z[(size_t)b * Z + j] = m;
    z[(size_t)b * Z + S + j] = fdec(smin[b * S + j]);
    z[(size_t)b * Z + 2 * S + j] = fdec(smax[b * S + j]);
    float var = (ssq[b * S + j] / c - m * m) * bessel;
    z[(size_t)b * Z + 3 * S + j] = sqrtf(fmaxf(var, 0.0f) + 1e-12f);
  }
  for (int i = 0; i < V; ++i) {
    float m = vns[b * V + i] / c;
    z[(size_t)b * Z + 4 * S + i] = fdec(vnmin[b * V + i]);
    z[(size_t)b * Z + 4 * S + V + i] = fdec(vnmax[b * V + i]);
    float var = (vnq[b * V + i] / c - m * m) * bessel;
    z[(size_t)b * Z + 4 * S + 2 * V + i] = sqrtf(fmaxf(var, 0.0f) + 1e-12f);
    vm[((size_t)b * V + i) * 3]     = vsum[(b * V + i) * 3] / c;
    vm[((size_t)b * V + i) * 3 + 1] = vsum[(b * V + i) * 3 + 1] / c;
    vm[((size_t)b * V + i) * 3 + 2] = vsum[(b * V + i) * 3 + 2] / c;
  }
}

__global__ void build_y_kernel(float* __restrict__ Ys, float* __restrict__ Yv,
                               const float* __restrict__ pxs, const float* __restrict__ pxv,
                               const float* __restrict__ pes, const float* __restrict__ pev,
                               const float* __restrict__ yss, const float* __restrict__ ysv) {
  int b = blockIdx.x * blockDim.x + threadIdx.x;
  if (b >= Bb) return;
  for (int j = 0; j < 64; ++j) Ys[b * 128 + j] = pxs[b * 64 + j];
  for (int j = 0; j < 32; ++j) { Ys[b * 128 + 64 + j] = pes[b * 32 + j];
                                 Ys[b * 128 + 96 + j] = yss[b * 32 + j]; }
  for (int j = 0; j < 24; ++j) Yv[b * 48 + j] = pxv[b * 24 + j];
  for (int j = 0; j < 12; ++j) { Yv[b * 48 + 24 + j] = pev[b * 12 + j];
                                 Yv[b * 48 + 36 + j] = ysv[b * 12 + j]; }
}

// =====================================================================
extern "C" void kernel_launch(void* const* d_in, const int* in_sizes, int n_in,
                              void* d_out, int out_size, void* d_ws, size_t ws_size,
                              hipStream_t stream) {
  (void)in_sizes; (void)n_in; (void)out_size; (void)ws_size;
  const float* s  = (const float*)d_in[0];
  const float* v  = (const float*)d_in[1];
  const int*   ei = (const int*)d_in[2];
  const int*   bm = (const int*)d_in[3];
  const float* es = (const float*)d_in[4];
  const float* ev = (const float*)d_in[5];
  const float* gs = (const float*)d_in[6];
  const float* gv = (const float*)d_in[7];
#define PP(k) ((const float*)d_in[(k)])
  enum { q_ws = 8, q_wv, k_ws, k_wv, vx_ws, vx_wv, ve_ws, ve_wv, b_ws,
         msg_wh, msg_ws, msg_bs, msg_wv, xout_wh, xout_ws, xout_bs, xout_wv,
         eout_wh, eout_ws, eout_bs, eout_wv, pnax_wh, pnax_ws, pnax_bs, pnax_wv,
         pnae_wh, pnae_ws, pnae_bs, pnae_wv, y_wh, y_ws, y_bs, y_wv,
         yout_wh, yout_ws, yout_bs, yout_wv,
         xn_g, xn_b, xn_vw, en_g, en_b, en_vw, yn_g, yn_b, yn_vw };

  float* w = (float*)d_ws;
  // ---- union region A: projections + message pipeline (Nn*336 + Ee*624 floats)
  float* qsB = w;
  float* ksB = qsB + (size_t)Nn * 64;
  float* vxsB = ksB + (size_t)Nn * 64;
  float* qvB = vxsB + (size_t)Nn * 64;
  float* kvB = qvB + (size_t)Nn * 48;
  float* vxvB = kvB + (size_t)Nn * 48;
  float* beB = vxvB + (size_t)Nn * 48;
  float* vesB = beB + (size_t)Ee * 32;
  float* vevB = vesB + (size_t)Ee * 64;
  float* mcat = vevB + (size_t)Ee * 48;
  float* mvp = mcat + (size_t)Ee * 160;
  float* vhm = mvp + (size_t)Ee * 96;
  float* mso = vhm + (size_t)Ee * 96;
  float* PCend = mso + (size_t)Ee * 128;
  // ---- overlay region B (after step seg_accum): edge-out (Ee*644 <= Nn*336+Ee*624)
  float* esB = w;
  float* evB = esB + (size_t)Ee * 216;
  float* vhe = evB + (size_t)Ee * 72;
  float* ecat = vhe + (size_t)Ee * 72;
  float* desB = ecat + (size_t)Ee * 240;
  float* devB = desB + (size_t)Ee * 32;
  // ---- persistent region
  float* p = PCend;
  float* at_s = p; p += (size_t)Ee * 8;
  float* at_v = p; p += (size_t)Ee * 8;
  float* at_e = p; p += (size_t)Ee * 8;
  float* at   = p; p += (size_t)Ee * 8;
  unsigned* lmax = (unsigned*)p; p += (size_t)Nn * 8;
  float* den = p; p += (size_t)Nn * 8;
  float* cnt = p; p += Nn;
  float* xin_s = p; p += (size_t)Nn * 160;
  float* xin_v = p; p += (size_t)Nn * 108;
  float* vhx = p; p += (size_t)Nn * 108;
  float* xcat = p; p += (size_t)Nn * 196;
  float* dxs = p; p += (size_t)Nn * 64;
  float* dxv = p; p += (size_t)Nn * 24;
  int* ebm = (int*)p; p += Ee;
  float* nc = p; p += Bb;
  float* nss = p; p += (size_t)Bb * 64;
  float* nsq = p; p += (size_t)Bb * 64;
  unsigned* nmin = (unsigned*)p; p += (size_t)Bb * 64;
  unsigned* nmax = (unsigned*)p; p += (size_t)Bb * 64;
  float* nvs = p; p += Bb * 8;
  float* nvq = p; p += Bb * 8;
  unsigned* nvmin = (unsigned*)p; p += Bb * 8;
  unsigned* nvmax = (unsigned*)p; p += Bb * 8;
  float* nvsum = p; p += Bb * 24;
  float* ec = p; p += Bb;
  float* ess_ = p; p += Bb * 32;
  float* esq = p; p += Bb * 32;
  unsigned* emin = (unsigned*)p; p += Bb * 32;
  unsigned* emax = (unsigned*)p; p += Bb * 32;
  float* evs = p; p += Bb * 4;
  float* evq = p; p += Bb * 4;
  unsigned* evmin = (unsigned*)p; p += Bb * 4;
  unsigned* evmax = (unsigned*)p; p += Bb * 4;
  float* evsum = p; p += Bb * 12;
  float* zx = p; p += Bb * 280;
  float* vmx = p; p += Bb * 24;
  float* vhpx = p; p += Bb * 24;
  float* catx = p; p += Bb * 288;
  float* pxs = p; p += Bb * 64;
  float* pxv = p; p += Bb * 24;
  float* ze = p; p += Bb * 140;
  float* vme = p; p += Bb * 12;
  float* vhpe = p; p += Bb * 12;
  float* cate = p; p += Bb * 144;
  float* pes = p; p += Bb * 32;
  float* pev = p; p += Bb * 12;
  float* vhy = p; p += Bb * 12;
  float* ycat = p; p += Bb * 36;
  float* yss = p; p += Bb * 32;
  float* ysv = p; p += Bb * 12;
  float* Ys = p; p += Bb * 128;
  float* Yv = p; p += Bb * 48;
  float* vhyo = p; p += Bb * 48;
  float* yocat = p; p += Bb * 144;
  float* dys = p; p += Bb * 32;
  float* dyv = p; p += Bb * 12;

  // output layout (flat concat of (xs_o, xv_o, es_o, ev_o, gs_o, gv_o))
  float* out = (float*)d_out;
  float* o_xs = out;
  float* o_xv = out + (size_t)Nn * 64;
  float* o_es = out + (size_t)Nn * 88;
  float* o_ev = o_es + (size_t)Ee * 32;
  float* o_gs = o_ev + (size_t)Ee * 12;
  float* o_gv = o_gs + (size_t)Bb * 32;

#define LAUNCH(kern, n, ...)                                                      \
  kern<<<dim3((unsigned)(((long long)(n) + 255) / 256)), dim3(256), 0, stream>>>( \
      __VA_ARGS__)

  auto lin = [&](float* o, const float* A, const float* W, const float* bias,
                 int M, int K, int Nout) {
    lin_wmma_kernel<<<dim3((unsigned)(M / 16), (unsigned)(Nout / 16)), dim3(32), 0,
                      stream>>>(o, A, W, bias, M, K, Nout);
  };
  auto vlin = [&](float* o, const float* X, const float* W, long long NNp, int O, int V) {
    vlin_wmma_kernel<<<dim3((unsigned)((3 * NNp) / 16), (unsigned)((O + 15) / 16)),
                       dim3(32), 0, stream>>>(o, X, W, O, V);
  };

  // ---- init accumulators
  LAUNCH(fill_u32_kernel, Nn * 8, lmax, ENC_NINF, (long long)Nn * 8);
  LAUNCH(fill_f32_kernel, Nn * 8, den, 0.f, (long long)Nn * 8);
  LAUNCH(fill_f32_kernel, Nn, cnt, 0.f, (long long)Nn);
  LAUNCH(fill_f32_kernel, Nn * 160, xin_s, 0.f, (long long)Nn * 160);
  LAUNCH(fill_f32_kernel, Nn * 108, xin_v, 0.f, (long long)Nn * 108);
  LAUNCH(fill_f32_kernel, Bb, nc, 0.f, Bb);
  LAUNCH(fill_f32_kernel, Bb * 128, nss, 0.f, Bb * 128);           // nss + nsq
  LAUNCH(fill_u32_kernel, Bb * 64, nmin, ENC_PINF, Bb * 64);
  LAUNCH(fill_u32_kernel, Bb * 64, nmax, ENC_NINF, Bb * 64);
  LAUNCH(fill_f32_kernel, Bb * 16, nvs, 0.f, Bb * 16);             // nvs + nvq
  LAUNCH(fill_u32_kernel, Bb * 8, nvmin, ENC_PINF, Bb * 8);
  LAUNCH(fill_u32_kernel, Bb * 8, nvmax, ENC_NINF, Bb * 8);
  LAUNCH(fill_f32_kernel, Bb * 24, nvsum, 0.f, Bb * 24);
  LAUNCH(fill_f32_kernel, Bb, ec, 0.f, Bb);
  LAUNCH(fill_f32_kernel, Bb * 64, ess_, 0.f, Bb * 64);            // ess + esq
  LAUNCH(fill_u32_kernel, Bb * 32, emin, ENC_PINF, Bb * 32);
  LAUNCH(fill_u32_kernel, Bb * 32, emax, ENC_NINF, Bb * 32);
  LAUNCH(fill_f32_kernel, Bb * 8, evs, 0.f, Bb * 8);               // evs + evq
  LAUNCH(fill_u32_kernel, Bb * 4, evmin, ENC_PINF, Bb * 4);
  LAUNCH(fill_u32_kernel, Bb * 4, evmax, ENC_NINF, Bb * 4);
  LAUNCH(fill_f32_kernel, Bb * 12, evsum, 0.f, Bb * 12);

  // ---- node / edge projections (WMMA)
  lin(qsB, s, PP(q_ws), nullptr, Nn, 64, 64);
  lin(ksB, s, PP(k_ws), nullptr, Nn, 64, 64);
  lin(vxsB, s, PP(vx_ws), nullptr, Nn, 64, 64);
  vlin(qvB, v, PP(q_wv), Nn, 16, 8);
  vlin(kvB, v, PP(k_wv), Nn, 16, 8);
  vlin(vxvB, v, PP(vx_wv), Nn, 16, 8);
  lin(beB, es, PP(b_ws), nullptr, Ee, 32, 32);
  lin(vesB, es, PP(ve_ws), nullptr, Ee, 32, 64);
  vlin(vevB, ev, PP(ve_wv), Ee, 16, 4);

  // ---- attention (segment softmax over rows)
  LAUNCH(attn_logits_kernel, Ee * Hh, ei, qsB, ksB, qvB, kvB, beB, at_s, at_v, at_e, at, lmax);
  LAUNCH(attn_exp_kernel, Ee * Hh, ei, at, lmax, den);
  LAUNCH(attn_norm_kernel, Ee * Hh, ei, at, den);
  LAUNCH(count_edges_kernel, Ee, ei, cnt);

  // ---- messages + msg GVP
  LAUNCH(build_msg_kernel, Ee * Hh, ei, at, vxsB, vesB, vxvB, vevB, mcat, mvp);
  vlin(vhm, mvp, PP(msg_wh), Ee, 32, 32);
  LAUNCH(concat_vn_kernel, (long long)Ee * 160, mcat, mcat, 128, vhm, 32, (long long)Ee);
  lin(mso, mcat, PP(msg_ws), PP(msg_bs), Ee, 160, 128);
  vlin(mvp, vhm, PP(msg_wv), Ee, 32, 32);  // mv_out overwrites mv_pre
  LAUNCH(seg_accum_kernel, (long long)Ee * 224, ei, mso, mvp, xin_s, xin_v);
  LAUNCH(finish_xin_kernel, Nn, bm, gs, gv, cnt, xin_s, xin_v);

  // ---- node update GVP + LN
  vlin(vhx, xin_v, PP(xout_wh), Nn, 36, 36);
  LAUNCH(concat_vn_kernel, (long long)Nn * 196, xcat, xin_s, 160, vhx, 36, (long long)Nn);
  lin(dxs, xcat, PP(xout_ws), PP(xout_bs), Nn, 196, 64);
  vlin(dxv, vhx, PP(xout_wv), Nn, 8, 36);
  LAUNCH(tuple_ln_kernel, Nn, o_xs, o_xv, s, dxs, v, dxv, PP(xn_g), PP(xn_b), PP(xn_vw),
         64, 8, (long long)Nn);

  // ---- edge update GVP + LN (reuses union region; message buffers are dead now)
  LAUNCH(ebm_kernel, Ee, ebm, ei, bm);
  LAUNCH(build_es_ev_kernel, (long long)Ee * 288, ei, ebm, s, v, es, ev, gs, gv,
         at_s, at_v, at_e, esB, evB);
  vlin(vhe, evB, PP(eout_wh), Ee, 24, 24);
  LAUNCH(concat_vn_kernel, (long long)Ee * 240, ecat, esB, 216, vhe, 24, (long long)Ee);
  lin(desB, ecat, PP(eout_ws), PP(eout_bs), Ee, 240, 32);
  vlin(devB, vhe, PP(eout_wv), Ee, 4, 24);
  LAUNCH(tuple_ln_kernel, Ee, o_es, o_ev, es, desB, ev, devB, PP(en_g), PP(en_b),
         PP(en_vw), 32, 4, (long long)Ee);

  // ---- PNA pooling (nodes & edges) + global update
  LAUNCH(pna_accum_kernel, Nn, s, v, bm, nc, nss, nsq, nmin, nmax, nvs, nvq, nvmin,
         nvmax, nvsum, 64, 8, Nn);
  LAUNCH(pna_accum_kernel, Ee, es, ev, ebm, ec, ess_, esq, emin, emax, evs, evq,
         evmin, evmax, evsum, 32, 4, Ee);
  LAUNCH(pna_final_kernel, Bb, zx, vmx, nc, nss, nsq, nmin, nmax, nvs, nvq, nvmin,
         nvmax, nvsum, 64, 8, Bb);
  LAUNCH(pna_final_kernel, Bb, ze, vme, ec, ess_, esq, emin, emax, evs, evq, evmin,
         evmax, evsum, 32, 4, Bb);
  vlin(vhpx, vmx, PP(pnax_wh), Bb, 8, 8);
  LAUNCH(concat_vn_kernel, (long long)Bb * 288, catx, zx, 280, vhpx, 8, (long long)Bb);
  lin(pxs, catx, PP(pnax_ws), PP(pnax_bs), Bb, 288, 64);
  vlin(pxv, vhpx, PP(pnax_wv), Bb, 8, 8);
  vlin(vhpe, vme, PP(pnae_wh), Bb, 4, 4);
  LAUNCH(concat_vn_kernel, (long long)Bb * 144, cate, ze, 140, vhpe, 4, (long long)Bb);
  lin(pes, cate, PP(pnae_ws), PP(pnae_bs), Bb, 144, 32);
  vlin(pev, vhpe, PP(pnae_wv), Bb, 4, 4);
  vlin(vhy, gv, PP(y_wh), Bb, 4, 4);
  LAUNCH(concat_vn_kernel, (long long)Bb * 36, ycat, gs, 32, vhy, 4, (long long)Bb);
  lin(yss, ycat, PP(y_ws), PP(y_bs), Bb, 36, 32);
  vlin(ysv, vhy, PP(y_wv), Bb, 4, 4);
  LAUNCH(build_y_kernel, Bb, Ys, Yv, pxs, pxv, pes, pev, yss, ysv);
  vlin(vhyo, Yv, PP(yout_wh), Bb, 16, 16);
  LAUNCH(concat_vn_kernel, (long long)Bb * 144, yocat, Ys, 128, vhyo, 16, (long long)Bb);
  lin(dys, yocat, PP(yout_ws), PP(yout_bs), Bb, 144, 32);
  vlin(dyv, vhyo, PP(yout_wv), Bb, 4, 16);
  LAUNCH(tuple_ln_kernel, Bb, o_gs, o_gv, gs, dys, gv, dyv, PP(yn_g), PP(yn_b),
         PP(yn_vw), 32, 4, (long long)Bb);
#undef LAUNCH
#undef PP
}